// VectorQuantizer_24026047054214
// MI455X (gfx1250) — compile-verified
//
#include <hip/hip_runtime.h>
#include <math.h>

// ---------------- problem constants ----------------
constexpr int Dn = 256;        // embedding_dim
constexpr int Kn = 1024;       // num_embeddings
constexpr int Nn = 65536;      // tokens
constexpr float FBETA  = 0.25f;
constexpr float FDECAY = 0.99f;
constexpr float FEPS   = 1e-5f;

typedef float v2f __attribute__((ext_vector_type(2)));
typedef float v8f __attribute__((ext_vector_type(8)));

// ---------------- d_out layout (floats, reference tuple order) ----------------
constexpr size_t Q_OFF    = 0;                          // quantized_st  [N,D]
constexpr size_t LOSS_OFF = (size_t)Nn * Dn;            // vq_loss       [1]
constexpr size_t IDX_OFF  = LOSS_OFF + 1;               // indices       [N]
constexpr size_t PERP_OFF = IDX_OFF + Nn;               // perplexity    [1]
constexpr size_t CS_OFF   = PERP_OFF + 1;               // new_cs        [K]
constexpr size_t EMAW_OFF = CS_OFF + Kn;                // new_ema_w     [K,D]
constexpr size_t W_OFF    = EMAW_OFF + (size_t)Kn * Dn; // new_weight    [K,D]

// ---------------- workspace layout (float units) ----------------
constexpr size_t WS_COUNTS = 0;                 // K floats
constexpr size_t WS_CNORM  = Kn;                // K floats
constexpr size_t WS_LOSS   = 2 * Kn;            // 1 float
constexpr size_t WS_IDX    = 4096;              // N ints
constexpr size_t WS_DW     = WS_IDX + Nn;       // K*D floats
constexpr size_t WS_TOTAL  = WS_DW + (size_t)Kn * Dn;

// ---------------- kernels ----------------
__global__ void k_init(float* __restrict__ ws) {
    size_t i = (size_t)blockIdx.x * blockDim.x + threadIdx.x;
    if (i < WS_TOTAL) ws[i] = 0.0f;
}

// ||codebook[k]||^2
__global__ __launch_bounds__(256) void k_cnorm(const float* __restrict__ cb,
                                               float* __restrict__ cnorm) {
    __shared__ float s[256];
    const int k = blockIdx.x;
    const float v = cb[(size_t)k * Dn + threadIdx.x];
    s[threadIdx.x] = v * v;
    __syncthreads();
    for (int st = 128; st > 0; st >>= 1) {
        if (threadIdx.x < st) s[threadIdx.x] += s[threadIdx.x + st];
        __syncthreads();
    }
    if (threadIdx.x == 0) cnorm[k] = s[0];
}

// Distance GEMM + row argmin via fp32 WMMA (v_wmma_f32_16x16x4_f32).
// Block: 256 threads = 8 waves; each wave owns a 16-token tile; all waves
// share a double-buffered LDS codebook chunk filled by async global->LDS
// copies (ASYNCcnt-tracked, no VGPR round-trip).
__global__ __launch_bounds__(256) void k_argmin(
    const float* __restrict__ x, const float* __restrict__ cb,
    const float* __restrict__ cnorm, int* __restrict__ idx_i,
    float* __restrict__ idx_f)
{
    __shared__ float lds_cb[2][16][260];  // 2 x (16 codes x 256 dims), padded rows
    __shared__ float lds_cn[Kn];          // all code norms

    const int tid   = threadIdx.x;
    const int wave  = tid >> 5;
    const int lane  = tid & 31;
    const int laneM = lane & 15;
    const int hi    = lane >> 4;          // which K-pair of the 16x4 fragment
    const int tokBase = blockIdx.x * 128 + wave * 16;

    for (int i = tid; i < Kn; i += 256) lds_cn[i] = cnorm[i];

    // Preload A fragments: 16 tokens x 256 dims per wave.
    // 16x4 f32 A layout: lanes 0-15 -> M=lane, K={0,1}; lanes 16-31 -> K={2,3}.
    const v2f* xr2 = reinterpret_cast<const v2f*>(x + (size_t)(tokBase + laneM) * Dn);
    v2f areg[64];
#pragma unroll
    for (int d4 = 0; d4 < 64; ++d4) areg[d4] = xr2[2 * d4 + hi];

    float runMin[8];
    int   runIdx[8];
#pragma unroll
    for (int r = 0; r < 8; ++r) { runMin[r] = 3.4e38f; runIdx[r] = 0; }

    // Async-stage one 16-code chunk (16KB) into LDS buffer `buf`:
    // 256 threads x 4 x b128 = 16KB, GVS mode (sgpr base + vgpr byte offset).
    auto stage = [&](int chunk, int buf) {
#pragma unroll
        for (int p = 0; p < 4; ++p) {
            const int i = (p * 256 + tid) * 4;          // float index in 16x256 tile
            const int r = i >> 8, c = i & 255;
            const unsigned ldsa = (unsigned)(size_t)&lds_cb[buf][r][c];
            const unsigned goff = (unsigned)(((chunk * 16 + r) * Dn + c) * 4);
            asm volatile("global_load_async_to_lds_b128 %0, %1, %2"
                         :: "v"(ldsa), "v"(goff), "s"(cb)
                         : "memory");
        }
    };

    stage(0, 0);
    for (int chunk = 0; chunk < Kn / 16; ++chunk) {
        const int buf = chunk & 1;
        if (chunk + 1 < Kn / 16) {
            stage(chunk + 1, buf ^ 1);
            // async loads complete in order: <=4 outstanding => chunk's 4 landed
            asm volatile("s_wait_asynccnt 0x4" ::: "memory");
        } else {
            asm volatile("s_wait_asynccnt 0x0" ::: "memory");
        }
        __syncthreads();

        // acc[m][n] = sum_d x[m][d] * cb[n][d]
        v8f acc = {};
#pragma unroll
        for (int d4 = 0; d4 < 64; ++d4) {
            // B (4x16): lane n holds column n (code n), same K-pair split as A.
            v2f b = *reinterpret_cast<const v2f*>(&lds_cb[buf][laneM][4 * d4 + 2 * hi]);
            acc = __builtin_amdgcn_wmma_f32_16x16x4_f32(
                    false, areg[d4], false, b, (short)0, acc, false, false);
        }

        // dist = ||c||^2 - 2 x.c   (||x||^2 constant per row -> drop for argmin)
        const float cn  = lds_cn[chunk * 16 + laneM];
        const int   col = chunk * 16 + laneM;
#pragma unroll
        for (int r = 0; r < 8; ++r) {
            const float dist = cn - 2.0f * acc[r];
            if (dist < runMin[r]) { runMin[r] = dist; runIdx[r] = col; }
        }
        __syncthreads();   // done reading buf before chunk+2's async copy overwrites it
    }

    // C/D layout: VGPR r, lanes 0-15 -> row r; lanes 16-31 -> row r+8.
    // Butterfly argmin across each 16-lane half (xor masks 1..8 never cross bit4).
#pragma unroll
    for (int r = 0; r < 8; ++r) {
        float m  = runMin[r];
        int   ix = runIdx[r];
#pragma unroll
        for (int off = 8; off > 0; off >>= 1) {
            const float om = __shfl_xor(m, off, 32);
            const int   oi = __shfl_xor(ix, off, 32);
            if (om < m || (om == m && oi < ix)) { m = om; ix = oi; }
        }
        if (laneM == 0) {
            const int t2 = tokBase + r + hi * 8;
            idx_i[t2] = ix;
            idx_f[t2] = (float)ix;
        }
    }
}

// Gather quantized rows, accumulate loss / counts / dw (segment sums).
__global__ __launch_bounds__(256) void k_scatter(
    const float* __restrict__ x, const float* __restrict__ cb,
    const int* __restrict__ idx, float* __restrict__ out_q,
    float* __restrict__ dw, float* __restrict__ counts,
    float* __restrict__ lossSum)
{
    __shared__ float s[256];
    const int t = blockIdx.x, d = threadIdx.x;
    const int ix = idx[t];
    const float q  = cb[(size_t)ix * Dn + d];
    const float xv = x[(size_t)t * Dn + d];
    out_q[(size_t)t * Dn + d] = q;          // straight-through value == quantized
    const float diff = q - xv;
    s[d] = diff * diff;
    atomicAdd(&dw[(size_t)ix * Dn + d], xv);
    __syncthreads();
    for (int st = 128; st > 0; st >>= 1) {
        if (d < st) s[d] += s[d + st];
        __syncthreads();
    }
    if (d == 0) {
        atomicAdd(lossSum, s[0]);
        atomicAdd(&counts[ix], 1.0f);
    }
}

// EMA cluster sizes + perplexity + loss finalization (single 1024-thread block).
__global__ __launch_bounds__(1024) void k_cs(
    const float* __restrict__ ema_cs, const float* __restrict__ counts,
    const float* __restrict__ lossSum, float* __restrict__ out_cs,
    float* __restrict__ out_perp, float* __restrict__ out_loss)
{
    __shared__ float sn[1024];
    __shared__ float se[1024];
    const int k = threadIdx.x;
    const float cnt = counts[k];
    const float raw = FDECAY * ema_cs[k] + (1.0f - FDECAY) * cnt;
    const float p   = cnt / (float)Nn;
    sn[k] = raw;
    se[k] = -p * logf(p + 1e-10f);
    __syncthreads();
    for (int st = 512; st > 0; st >>= 1) {
        if (k < st) { sn[k] += sn[k + st]; se[k] += se[k + st]; }
        __syncthreads();
    }
    const float ntot = sn[0];
    out_cs[k] = (raw + FEPS) / (ntot + (float)Kn * FEPS) * ntot;
    if (k == 0) {
        *out_perp = expf(se[0]);
        *out_loss = FBETA * (*lossSum) / (float)((size_t)Nn * Dn);
    }
}

// EMA weights and normalized codebook.
__global__ __launch_bounds__(256) void k_emaw(
    const float* __restrict__ ema_w, const float* __restrict__ dw,
    const float* __restrict__ new_cs, float* __restrict__ out_emaw,
    float* __restrict__ out_w)
{
    const size_t i = (size_t)blockIdx.x * 256 + threadIdx.x;
    const int k = (int)(i >> 8);
    const float e = FDECAY * ema_w[i] + (1.0f - FDECAY) * dw[i];
    out_emaw[i] = e;
    out_w[i]    = e / new_cs[k];
}

// ---------------- launcher ----------------
extern "C" void kernel_launch(void* const* d_in, const int* in_sizes, int n_in,
                              void* d_out, int out_size, void* d_ws, size_t ws_size,
                              hipStream_t stream) {
    const float* x      = (const float*)d_in[0];  // [N,D]
    const float* cb     = (const float*)d_in[1];  // [K,D]
    const float* ema_cs = (const float*)d_in[2];  // [K]
    const float* ema_w  = (const float*)d_in[3];  // [K,D]

    float* out = (float*)d_out;
    float* ws  = (float*)d_ws;

    float* ws_counts = ws + WS_COUNTS;
    float* ws_cnorm  = ws + WS_CNORM;
    float* ws_loss   = ws + WS_LOSS;
    int*   ws_idx    = (int*)(ws + WS_IDX);
    float* ws_dw     = ws + WS_DW;

    k_init  <<<(int)((WS_TOTAL + 255) / 256), 256, 0, stream>>>(ws);
    k_cnorm <<<Kn, 256, 0, stream>>>(cb, ws_cnorm);
    k_argmin<<<Nn / 128, 256, 0, stream>>>(x, cb, ws_cnorm, ws_idx, out + IDX_OFF);
    k_scatter<<<Nn, 256, 0, stream>>>(x, cb, ws_idx, out + Q_OFF, ws_dw,
                                      ws_counts, ws_loss);
    k_cs    <<<1, 1024, 0, stream>>>(ema_cs, ws_counts, ws_loss,
                                     out + CS_OFF, out + PERP_OFF, out + LOSS_OFF);
    k_emaw  <<<(Kn * Dn) / 256, 256, 0, stream>>>(ema_w, ws_dw, out + CS_OFF,
                                                  out + EMAW_OFF, out + W_OFF);
}